// _MSA2_61203283968820
// MI455X (gfx1250) — compile-verified
//
#include <hip/hip_runtime.h>

typedef __attribute__((ext_vector_type(2))) float v2f;
typedef __attribute__((ext_vector_type(8))) float v8f;

// Problem constants (from reference): B=2, nhid=100, N=20000, L=16, O=100
#define NB   2
#define NC   100     // nhid == O == 100
#define NN   20000
#define NL   16
#define MPAD 112     // 7 tiles of 16 rows (output-channel dim padded)
#define NTILE 16

// ---------------------------------------------------------------------------
// Kernel 1: Weff[112][100] = Wo(100x100) @ Wv(100x100), zero-padded rows;
//           beff[112]      = 16 * Wo @ bv + bo, zero-padded.
// ---------------------------------------------------------------------------
__global__ __launch_bounds__(256) void _MSA2_prep(
    const float* __restrict__ Wv, const float* __restrict__ bv,
    const float* __restrict__ Wo, const float* __restrict__ bo,
    float* __restrict__ Weff, float* __restrict__ beff)
{
    int idx = blockIdx.x * 256 + threadIdx.x;
    if (idx < MPAD * NC) {
        int o  = idx / NC;
        int ch = idx % NC;
        float s = 0.f;
        if (o < NC) {
            #pragma unroll 4
            for (int c = 0; c < NC; ++c)
                s = fmaf(Wo[o * NC + c], Wv[c * NC + ch], s);
        }
        Weff[idx] = s;
    } else if (idx < MPAD * NC + MPAD) {
        int o = idx - MPAD * NC;
        float s = 0.f;
        if (o < NC) {
            float t = 0.f;
            #pragma unroll 4
            for (int c = 0; c < NC; ++c)
                t = fmaf(Wo[o * NC + c], bv[c], t);
            s = 16.f * t + bo[o];
        }
        beff[o] = s;
    }
}

// ---------------------------------------------------------------------------
// Kernel 2 (fused, memory-bound): per block = one (b, 16-column n-tile).
//   Phase 1: stream y[b, c, n0..n0+15, 0..15], reduce L in registers,
//            deposit ysum tile (100 x 16 fp32) in LDS.
//   Phase 2: 7 waves x 25 steps of V_WMMA_F32_16X16X4_F32 computing
//            out_tile(112x16) = Weff(112x100) @ ysum(100x16), + beff.
// ---------------------------------------------------------------------------
__global__ __launch_bounds__(256) void _MSA2_main(
    const float* __restrict__ y,
    const float* __restrict__ Weff,
    const float* __restrict__ beff,
    float* __restrict__ out)
{
    __shared__ float lds[NC * NTILE];   // ysum tile: [k=0..99][nn=0..15]

    const int n0  = blockIdx.x * NTILE;
    const int b   = blockIdx.y;
    const int tid = threadIdx.x;

    // ---- Phase 1: coalesced stream of y + L-reduction ----
    const float* yb = y + (size_t)b * NC * NN * NL;
    for (int idx = tid; idx < NC * NTILE; idx += 256) {
        const int c  = idx >> 4;
        const int nn = idx & 15;
        const float4* p =
            (const float4*)(yb + ((size_t)c * NN + (size_t)(n0 + nn)) * NL);
        float4 a0 = p[0], a1 = p[1], a2 = p[2], a3 = p[3];
        float s = ((a0.x + a0.y) + (a0.z + a0.w))
                + ((a1.x + a1.y) + (a1.z + a1.w))
                + ((a2.x + a2.y) + (a2.z + a2.w))
                + ((a3.x + a3.y) + (a3.z + a3.w));
        lds[idx] = s;
    }
    __syncthreads();

    // ---- Phase 2: fp32 WMMA GEMM (wave-uniform branch; EXEC all-ones) ----
    const int wave = tid >> 5;
    const int lane = tid & 31;
    if (wave < 7) {
        const int m0   = wave * 16;
        const int half = lane >> 4;      // 0: K pair {0,1}; 1: K pair {2,3}
        const int lr   = lane & 15;      // A: row m0+lr ; B/D: column lr

        // A row pointer: Weff[(m0+lr)][2*half + k0 ...], 8B-aligned
        const float2* Arow =
            (const float2*)(Weff + (size_t)(m0 + lr) * NC + 2 * half);

        v8f acc = {};
        for (int k0 = 0; k0 < NC; k0 += 4) {   // 25 WMMA steps, K=100 exact
            float2 av = Arow[k0 >> 1];
            v2f a; a.x = av.x; a.y = av.y;

            const int kb = k0 + 2 * half;
            v2f bf;
            bf.x = lds[kb * NTILE + lr];
            bf.y = lds[(kb + 1) * NTILE + lr];

            // (neg_a, A, neg_b, B, c_mod, C, reuse_a, reuse_b)
            acc = __builtin_amdgcn_wmma_f32_16x16x4_f32(
                false, a, false, bf, (short)0, acc, false, false);
        }

        // D layout: VGPR i -> row m0+i (lanes 0-15) / m0+i+8 (lanes 16-31)
        float* ob = out + (size_t)b * NC * NN;
        const int col = n0 + lr;
        #pragma unroll
        for (int i = 0; i < 8; ++i) {
            const int row = m0 + i + half * 8;
            if (row < NC)
                ob[(size_t)row * NN + col] = acc[i] + beff[row];
        }
    }
}

extern "C" void kernel_launch(void* const* d_in, const int* in_sizes, int n_in,
                              void* d_out, int out_size, void* d_ws, size_t ws_size,
                              hipStream_t stream) {
    // Inputs (setup_inputs order): x, y, Wq, bq, Wk, bk, Wv, bv, Wo, bo
    const float* y  = (const float*)d_in[1];
    const float* Wv = (const float*)d_in[6];
    const float* bv = (const float*)d_in[7];
    const float* Wo = (const float*)d_in[8];
    const float* bo = (const float*)d_in[9];
    float* out = (float*)d_out;

    float* Weff = (float*)d_ws;          // 112*100 floats
    float* beff = Weff + MPAD * NC;      // 112 floats

    // Kernel 1: fold Wo@Wv and the bias chain (softmax(axis=1)==1 makes
    // q/k/x dead; the two torch .view scrambles cancel exactly).
    const int prep_items = MPAD * NC + MPAD;
    _MSA2_prep<<<(prep_items + 255) / 256, 256, 0, stream>>>(
        Wv, bv, Wo, bo, Weff, beff);

    // Kernel 2: 1250 n-tiles x 2 batches, 8 waves per block.
    dim3 grid(NN / NTILE, NB);
    _MSA2_main<<<grid, 256, 0, stream>>>(y, Weff, beff, out);
    (void)in_sizes; (void)n_in; (void)out_size; (void)ws_size;
}